// SAGEGNN_6691559047585
// MI455X (gfx1250) — compile-verified
//
#include <hip/hip_runtime.h>
#include <hip/hip_bf16.h>
#include <stdint.h>

typedef float v2f __attribute__((ext_vector_type(2)));
typedef float v8f __attribute__((ext_vector_type(8)));

#define N_NODES 10000
#define D 128
#define N_EDGES 640000

// ---------------------------------------------------------------------------
// 1) Zero the aggregation buffer and degree counter in workspace.
// ---------------------------------------------------------------------------
__global__ void sage_init_ws(float* __restrict__ agg, float* __restrict__ deg) {
    int t = blockIdx.x * blockDim.x + threadIdx.x;
    int stride = gridDim.x * blockDim.x;
    for (int i = t; i < N_NODES * D; i += stride) agg[i] = 0.0f;
    for (int i = t; i < N_NODES; i += stride) deg[i] = 0.0f;
}

// ---------------------------------------------------------------------------
// 2) Edge-parallel scatter-add: one wave32 per edge, one float4 per lane.
//    x (5.12 MB) and agg (5.12 MB) both live in the 192 MB L2, so the
//    gather hits cache and the f32 atomics resolve at L2.
// ---------------------------------------------------------------------------
__global__ void sage_scatter(const float* __restrict__ x,
                             const long long* __restrict__ ei,   // [2, E] int64
                             float* __restrict__ agg,
                             float* __restrict__ deg) {
    long long t = (long long)blockIdx.x * blockDim.x + threadIdx.x;
    int e    = (int)(t >> 5);
    int lane = (int)(t & 31);
    if (e >= N_EDGES) return;
    int src = (int)ei[e];
    int dst = (int)ei[N_EDGES + e];

    const float4 v = *(const float4*)(x + (size_t)src * D + lane * 4);
    float* a = agg + (size_t)dst * D + lane * 4;
    __hip_atomic_fetch_add(a + 0, v.x, __ATOMIC_RELAXED, __HIP_MEMORY_SCOPE_AGENT);
    __hip_atomic_fetch_add(a + 1, v.y, __ATOMIC_RELAXED, __HIP_MEMORY_SCOPE_AGENT);
    __hip_atomic_fetch_add(a + 2, v.z, __ATOMIC_RELAXED, __HIP_MEMORY_SCOPE_AGENT);
    __hip_atomic_fetch_add(a + 3, v.w, __ATOMIC_RELAXED, __HIP_MEMORY_SCOPE_AGENT);
    if (lane == 0)
        __hip_atomic_fetch_add(deg + dst, 1.0f, __ATOMIC_RELAXED, __HIP_MEMORY_SCOPE_AGENT);
}

// ---------------------------------------------------------------------------
// 3) Mean-normalize in place: agg[i,:] *= (deg[i] > 0 ? 1/deg[i] : 0).
// ---------------------------------------------------------------------------
__global__ void sage_normalize(float* __restrict__ agg, const float* __restrict__ deg) {
    int t = blockIdx.x * blockDim.x + threadIdx.x;   // one float4 group per thread
    if (t >= N_NODES * (D / 4)) return;
    int node = t >> 5;                               // 32 groups of 4 per node
    float d = deg[node];
    float s = (d > 0.0f) ? (1.0f / d) : 0.0f;
    float4* p = (float4*)agg + t;
    float4 v = *p;
    v.x *= s; v.y *= s; v.z *= s; v.w *= s;
    *p = v;
}

// ---------------------------------------------------------------------------
// 4) Fused dual-GEMM + bias + residual via V_WMMA_F32_16X16X4_F32.
//    One wave computes one 16x16 output tile:
//      acc += A_agg(16x4) x W_l^T(4x16)  and  acc += A_x(16x4) x W_r^T(4x16)
//    over K = 128 in steps of 4. 10000 = 625*16, 128 = 8*16 -> no bounds
//    checks, EXEC stays all-ones (WMMA requirement).
//    f32 A 16x4 layout: lane l holds (m = l&15, k = 2*(l>>4), 2*(l>>4)+1)
//    f32 B 4x16 layout: lane l holds (n = l&15, same k pair); B[k][n] = W[n][k]
//    C/D 16x16 layout: lane l, VGPR v -> (m = v + 8*(l>>4), n = l&15)
// ---------------------------------------------------------------------------
__global__ void __launch_bounds__(256)
sage_gemm(const float* __restrict__ x, const float* __restrict__ aggm,
          const float* __restrict__ Wl, const float* __restrict__ bl,
          const float* __restrict__ Wr, float* __restrict__ out) {
    int wave = threadIdx.x >> 5;
    int lane = threadIdx.x & 31;
    int tile = blockIdx.x * 8 + wave;        // 625 blocks * 8 waves = 5000 tiles
    int row0 = (tile >> 3) << 4;             // 0..9984
    int col0 = (tile & 7) << 4;              // 0..112
    int r  = lane & 15;
    int kh = (lane >> 4) << 1;               // k sub-offset: 0 or 2

    const float* aP = aggm + (size_t)(row0 + r) * D + kh;  // A rows (agg path)
    const float* xP = x    + (size_t)(row0 + r) * D + kh;  // A rows (root path)
    const float* lP = Wl   + (size_t)(col0 + r) * D + kh;  // B cols = W_l rows
    const float* rP = Wr   + (size_t)(col0 + r) * D + kh;  // B cols = W_r rows

    v8f acc = {};
#pragma unroll
    for (int kk = 0; kk < D; kk += 4) {
        v2f aL = *(const v2f*)(aP + kk);
        v2f bL = *(const v2f*)(lP + kk);
        v2f aR = *(const v2f*)(xP + kk);
        v2f bR = *(const v2f*)(rP + kk);
        acc = __builtin_amdgcn_wmma_f32_16x16x4_f32(
                  false, aL, false, bL, (short)0, acc, false, false);
        acc = __builtin_amdgcn_wmma_f32_16x16x4_f32(
                  false, aR, false, bR, (short)0, acc, false, false);
    }

    // epilogue: out = acc + b_l[n] + x (residual)
    float bln = bl[col0 + (lane & 15)];
    int mb = (lane >> 4) << 3;
#pragma unroll
    for (int v = 0; v < 8; ++v) {
        int m = mb + v;
        size_t idx = (size_t)(row0 + m) * D + col0 + (lane & 15);
        out[idx] = acc[v] + bln + x[idx];
    }
}

// ---------------------------------------------------------------------------
extern "C" void kernel_launch(void* const* d_in, const int* in_sizes, int n_in,
                              void* d_out, int out_size, void* d_ws, size_t ws_size,
                              hipStream_t stream) {
    const float*     x  = (const float*)d_in[0];
    const long long* ei = (const long long*)d_in[1];   // int64 edge_index [2, E]
    const float*     Wl = (const float*)d_in[2];
    const float*     bl = (const float*)d_in[3];
    const float*     Wr = (const float*)d_in[4];
    float* out = (float*)d_out;

    float* agg = (float*)d_ws;                 // N_NODES * D floats (5.12 MB)
    float* deg = agg + (size_t)N_NODES * D;    // N_NODES floats

    sage_init_ws<<<(N_NODES * D + 255) / 256, 256, 0, stream>>>(agg, deg);
    sage_scatter<<<(N_EDGES * 32) / 256, 256, 0, stream>>>(x, ei, agg, deg);
    sage_normalize<<<(N_NODES * (D / 4) + 255) / 256, 256, 0, stream>>>(agg, deg);
    sage_gemm<<<N_NODES / 16, 256, 0, stream>>>(x, agg, Wl, bl, Wr, out);
}